// GATEncoder_15685220565798
// MI455X (gfx1250) — compile-verified
//
#include <hip/hip_runtime.h>
#include <cstdint>
#include <cstddef>

// ---------------------------------------------------------------------------
// GAT layer for MI455X (gfx1250, wave32).
//   h = x@W via V_WMMA_F32_16X16X4_F32 (fp32 matrix pipe, exact)
//   segment softmax via ordered-uint atomicMax + f32 atomics (L2-resident)
// ---------------------------------------------------------------------------

typedef __attribute__((ext_vector_type(2))) float v2f;
typedef __attribute__((ext_vector_type(4))) float v4f;
typedef __attribute__((ext_vector_type(8))) float v8f;

#define NEG_SLOPE 0.2f

__device__ __forceinline__ float leaky(float x) { return x > 0.f ? x : NEG_SLOPE * x; }

// Monotone order-preserving mapping float -> uint32 so atomicMax(u32) == float max.
__device__ __forceinline__ unsigned fkey(float x) {
    unsigned b = __float_as_uint(x);
    return (b & 0x80000000u) ? ~b : (b | 0x80000000u);
}
__device__ __forceinline__ float funkey(unsigned k) {
    unsigned b = (k & 0x80000000u) ? (k ^ 0x80000000u) : ~k;
    return __uint_as_float(b);
}

// ---------------------------------------------------------------------------
// K1: h[N,32] = x[N,128] @ W[128,32] with V_WMMA_F32_16X16X4_F32.
// One wave computes a 16x32 tile of h (two 16x16 accumulators, 32 k-steps).
// W is staged in LDS pre-swizzled into B-fragment order:
//   sB[64*j + 32*half + 16*tile + l16] = { W[(4j+2*half)  *32 + 16*tile + l16],
//                                          W[(4j+2*half+1)*32 + 16*tile + l16] }
// A fragment (ISA 7.12.2, 32-bit A 16x4): lane l16 holds row M=l16;
//   lanes 0-15: V0=K(4j+0) V1=K(4j+1); lanes 16-31: V0=K(4j+2) V1=K(4j+3).
// Row index is CLAMPED (not exec-guarded) so the A loads are branchless and
// the scheduler can pipeline loads against WMMAs; stores stay guarded.
// ---------------------------------------------------------------------------
__global__ __launch_bounds__(256) void gat_gemm_wmma(const float* __restrict__ x,
                                                     const float* __restrict__ W,
                                                     float* __restrict__ h, int N) {
    __shared__ v2f sB[2048];  // 16 KB: [j=32][half=2][tile=2][l16=16]
    const int tid = threadIdx.x;
#pragma unroll
    for (int i = 0; i < 8; ++i) {
        int idx = tid + i * 256;  // 0..2047
        int l16 = idx & 15;
        int t   = (idx >> 4) & 1;
        int hf  = (idx >> 5) & 1;
        int j   = idx >> 6;
        int k   = 4 * j + 2 * hf;
        int col = 16 * t + l16;
        sB[idx] = v2f{W[k * 32 + col], W[(k + 1) * 32 + col]};
    }
    __syncthreads();

    const int lane = tid & 31;
    const int wave = tid >> 5;
    const int l16  = lane & 15;
    const int hf   = lane >> 4;  // lane half selects K pair
    const int row_base = (blockIdx.x * 8 + wave) * 16;
    if (row_base >= N) return;  // wave-uniform early out

    int arow = row_base + l16;
    arow = (arow < N) ? arow : (N - 1);  // clamp: always-legal branchless loads
    const v2f* __restrict__ x2   = (const v2f*)x;
    const v2f* __restrict__ xrow = x2 + (long)arow * 64 + hf;  // a(j) = xrow[2j]
    const v2f* __restrict__ bp   = sB + 32 * hf + l16;         // b(j,t) = bp[64j+16t]

    v8f acc0 = {};  // cols 0..15
    v8f acc1 = {};  // cols 16..31

    // one-step software pipeline: prefetch j+1 while WMMAing j
    v2f a  = xrow[0];
    v2f b0 = bp[0];
    v2f b1 = bp[16];
#pragma unroll
    for (int j = 0; j < 32; ++j) {
        v2f an = a, bn0 = b0, bn1 = b1;
        if (j < 31) {
            an  = xrow[2 * (j + 1)];
            bn0 = bp[64 * (j + 1)];
            bn1 = bp[64 * (j + 1) + 16];
        }
        acc0 = __builtin_amdgcn_wmma_f32_16x16x4_f32(false, a, false, b0, (short)0, acc0,
                                                     false, false);
        acc1 = __builtin_amdgcn_wmma_f32_16x16x4_f32(false, a, false, b1, (short)0, acc1,
                                                     false, false);
        a = an; b0 = bn0; b1 = bn1;
    }
    // D layout: VGPR r, lanes 0-15 -> M=r, lanes 16-31 -> M=r+8; N=l16.
#pragma unroll
    for (int r = 0; r < 8; ++r) {
        int row = row_base + r + 8 * hf;
        if (row < N) {
            h[(size_t)row * 32 + l16]      = acc0[r];
            h[(size_t)row * 32 + 16 + l16] = acc1[r];
        }
    }
}

// ---------------------------------------------------------------------------
// K2: per-node attention logits; init running max with self-loop logit.
// ---------------------------------------------------------------------------
__global__ __launch_bounds__(256) void gat_node_prep(const float* __restrict__ h,
                                                     const float* __restrict__ att_src,
                                                     const float* __restrict__ att_dst,
                                                     float* __restrict__ a_s,
                                                     float* __restrict__ a_d,
                                                     unsigned* __restrict__ mkey, int N) {
    int i = blockIdx.x * blockDim.x + threadIdx.x;
    if (i >= N) return;
    const v4f* h4  = (const v4f*)(h + (size_t)i * 32);
    const v4f* as4 = (const v4f*)att_src;
    const v4f* ad4 = (const v4f*)att_dst;
    float s = 0.f, d = 0.f;
#pragma unroll
    for (int j = 0; j < 8; ++j) {
        v4f hv = h4[j], av = as4[j], bv = ad4[j];
        s += hv.x * av.x + hv.y * av.y + hv.z * av.z + hv.w * av.w;
        d += hv.x * bv.x + hv.y * bv.y + hv.z * bv.z + hv.w * bv.w;
    }
    a_s[i] = s;
    a_d[i] = d;
    mkey[i] = fkey(leaky(s + d));  // self-loop edge (i,i) always exists
}

// ---------------------------------------------------------------------------
// K3: per-edge segment max (ordered-uint atomicMax -> global_atomic_max_u32).
// ---------------------------------------------------------------------------
__global__ __launch_bounds__(256) void gat_edge_max(const long long* __restrict__ ei,
                                                    const float* __restrict__ a_s,
                                                    const float* __restrict__ a_d,
                                                    unsigned* __restrict__ mkey, int E) {
    int e = blockIdx.x * blockDim.x + threadIdx.x;
    if (e >= E) return;
    int s = (int)ei[e];
    int d = (int)ei[(size_t)E + e];
    atomicMax(&mkey[d], fkey(leaky(a_s[s] + a_d[d])));
}

// ---------------------------------------------------------------------------
// K4: decode max, seed den and out (= unnormalized numerator) with self-loop.
// ---------------------------------------------------------------------------
__global__ __launch_bounds__(256) void gat_node_init(const float* __restrict__ h,
                                                     const float* __restrict__ a_s,
                                                     const float* __restrict__ a_d,
                                                     const unsigned* __restrict__ mkey,
                                                     float* __restrict__ mf,
                                                     float* __restrict__ den,
                                                     float* __restrict__ out, int N) {
    int i = blockIdx.x * blockDim.x + threadIdx.x;
    if (i >= N) return;
    float m = funkey(mkey[i]);
    float w = __expf(leaky(a_s[i] + a_d[i]) - m);
    mf[i]  = m;
    den[i] = w;
    const v4f* h4 = (const v4f*)(h + (size_t)i * 32);
    v4f* o4 = (v4f*)(out + (size_t)i * 32);
#pragma unroll
    for (int j = 0; j < 8; ++j) {
        v4f hv = h4[j];
        o4[j] = v4f{w * hv.x, w * hv.y, w * hv.z, w * hv.w};
    }
}

// ---------------------------------------------------------------------------
// K5: per-edge weighted scatter-add. 8 lanes per edge -> the h[src] gather is
// a coalesced 128B (8 x float4) line; atomics land in L2 (out fits in 192MB).
// ---------------------------------------------------------------------------
__global__ __launch_bounds__(256) void gat_edge_acc(const long long* __restrict__ ei,
                                                    const float* __restrict__ a_s,
                                                    const float* __restrict__ a_d,
                                                    const float* __restrict__ mf,
                                                    const float* __restrict__ h,
                                                    float* __restrict__ den,
                                                    float* __restrict__ out, int E) {
    long t  = (long)blockIdx.x * blockDim.x + threadIdx.x;
    long e  = t >> 3;
    int sub = (int)(t & 7);
    if (e >= E) return;
    int s = (int)ei[e];
    int d = (int)ei[(size_t)E + e];
    float w = __expf(leaky(a_s[s] + a_d[d]) - mf[d]);
    if (sub == 0) atomicAdd(&den[d], w);
    v4f hv = ((const v4f*)(h + (size_t)s * 32))[sub];
    float* op = out + (size_t)d * 32 + sub * 4;
    atomicAdd(op + 0, w * hv.x);
    atomicAdd(op + 1, w * hv.y);
    atomicAdd(op + 2, w * hv.z);
    atomicAdd(op + 3, w * hv.w);
}

// ---------------------------------------------------------------------------
// K6: out = sigmoid(out/den + bias), in place.
// ---------------------------------------------------------------------------
__global__ __launch_bounds__(256) void gat_finalize(float* __restrict__ out,
                                                    const float* __restrict__ den,
                                                    const float* __restrict__ bias,
                                                    long total) {
    long t = (long)blockIdx.x * blockDim.x + threadIdx.x;
    if (t >= total) return;
    int  c = (int)(t & 31);
    long i = t >> 5;
    float v = out[t] / den[i] + bias[c];
    out[t] = 1.f / (1.f + __expf(-v));
}

// ---------------------------------------------------------------------------
extern "C" void kernel_launch(void* const* d_in, const int* in_sizes, int n_in,
                              void* d_out, int out_size, void* d_ws, size_t ws_size,
                              hipStream_t stream) {
    const float*     x       = (const float*)d_in[0];
    const long long* ei      = (const long long*)d_in[1];  // int64 [2,E]
    const float*     W       = (const float*)d_in[2];
    const float*     att_src = (const float*)d_in[3];
    const float*     att_dst = (const float*)d_in[4];
    const float*     bias    = (const float*)d_in[5];
    const int N = in_sizes[0] / 128;
    const int E = in_sizes[1] / 2;
    (void)n_in; (void)out_size; (void)ws_size;

    char* ws = (char*)d_ws;
    auto alignup = [](size_t v) { return (v + 255) & ~(size_t)255; };
    size_t off = 0;
    float*    h    = (float*)(ws + off); off = alignup(off + (size_t)N * 32 * sizeof(float));
    float*    a_s  = (float*)(ws + off); off = alignup(off + (size_t)N * sizeof(float));
    float*    a_d  = (float*)(ws + off); off = alignup(off + (size_t)N * sizeof(float));
    unsigned* mkey = (unsigned*)(ws + off); off = alignup(off + (size_t)N * sizeof(unsigned));
    float*    mf   = (float*)(ws + off); off = alignup(off + (size_t)N * sizeof(float));
    float*    den  = (float*)(ws + off); off = alignup(off + (size_t)N * sizeof(float));
    float*    out  = (float*)d_out;  // accumulate numerator directly in d_out

    const int waves   = (N + 15) / 16;        // 16 rows per wave
    const int gemm_gb = (waves + 7) / 8;      // 8 waves per block
    const int node_gb = (N + 255) / 256;
    const int edge_gb = (E + 255) / 256;
    const long acc_t  = (long)E * 8;
    const int acc_gb  = (int)((acc_t + 255) / 256);
    const long fin_t  = (long)N * 32;
    const int fin_gb  = (int)((fin_t + 255) / 256);

    gat_gemm_wmma<<<gemm_gb, 256, 0, stream>>>(x, W, h, N);
    gat_node_prep<<<node_gb, 256, 0, stream>>>(h, att_src, att_dst, a_s, a_d, mkey, N);
    gat_edge_max<<<edge_gb, 256, 0, stream>>>(ei, a_s, a_d, mkey, E);
    gat_node_init<<<node_gb, 256, 0, stream>>>(h, a_s, a_d, mkey, mf, den, out, N);
    gat_edge_acc<<<acc_gb, 256, 0, stream>>>(ei, a_s, a_d, mf, h, den, out, E);
    gat_finalize<<<fin_gb, 256, 0, stream>>>(out, den, bias, fin_t);
}